// listlossmam_57612691309203
// MI455X (gfx1250) — compile-verified
//
#include <hip/hip_runtime.h>
#include <math.h>

typedef __attribute__((ext_vector_type(2))) float v2f;
typedef __attribute__((ext_vector_type(8))) float v8f;

#define RESCALE_ROT 1.0f
#define WEIGHT_PC   0.5f

// Workspace layout (floats):
//   ws[0]              : global error accumulator (sum of per-point errors)
//   ws[1]              : loss_rot
//   ws[16 + b*16 + k]  : per-batch (M - I) 3x3 row-major, M = Rp^T * Rt

__device__ inline void quat_to_rot3(float w, float x, float y, float z, float R[9]) {
    float inv = rsqrtf(w * w + x * x + y * y + z * z);
    w *= inv; x *= inv; y *= inv; z *= inv;
    R[0] = 1.f - 2.f * (y * y + z * z); R[1] = 2.f * (x * y - z * w);       R[2] = 2.f * (x * z + y * w);
    R[3] = 2.f * (x * y + z * w);       R[4] = 1.f - 2.f * (x * x + z * z); R[5] = 2.f * (y * z - x * w);
    R[6] = 2.f * (x * z - y * w);       R[7] = 2.f * (y * z + x * w);       R[8] = 1.f - 2.f * (x * x + y * y);
}

// One wave32: per-lane batch. Quaternion distance mean + (M - I) matrices.
__global__ void prep_kernel(const float* __restrict__ target_rot,
                            const float* __restrict__ rot_err,
                            float* __restrict__ ws, int B) {
    int b = threadIdx.x;
    float d = 0.0f;
    if (b < B) {
        float r0 = target_rot[b * 4 + 0], r1 = target_rot[b * 4 + 1];
        float r2 = target_rot[b * 4 + 2], r3 = target_rot[b * 4 + 3];
        float q0 = rot_err[b * 4 + 0], q1 = rot_err[b * 4 + 1];
        float q2 = rot_err[b * 4 + 2], q3 = rot_err[b * 4 + 3];

        // quaternion_distance(q = rot_err, r = target_rot): diff = q * conj(r)
        float c0 = r0, c1 = -r1, c2 = -r2, c3 = -r3;
        float t0 = c0 * q0 - c1 * q1 - c2 * q2 - c3 * q3;
        float t1 = c0 * q1 + c1 * q0 - c2 * q3 + c3 * q2;
        float t2 = c0 * q2 + c1 * q3 + c2 * q0 - c3 * q1;
        float t3 = c0 * q3 - c1 * q2 + c2 * q1 + c3 * q0;
        float vn = sqrtf(t1 * t1 + t2 * t2 + t3 * t3);
        d = 2.0f * atan2f(vn, fabsf(t0));

        // M = Rp^T * Rt (translation cancels out of inv(RT_total)); store M - I
        float Rt[9], Rp[9];
        quat_to_rot3(r0, r1, r2, r3, Rt);
        quat_to_rot3(q0, q1, q2, q3, Rp);
        float* mdst = ws + 16 + b * 16;
#pragma unroll
        for (int i = 0; i < 3; ++i)
#pragma unroll
            for (int j = 0; j < 3; ++j) {
                float m = Rp[0 * 3 + i] * Rt[0 * 3 + j]
                        + Rp[1 * 3 + i] * Rt[1 * 3 + j]
                        + Rp[2 * 3 + i] * Rt[2 * 3 + j];
                mdst[i * 3 + j] = m - ((i == j) ? 1.0f : 0.0f);
            }
    }
#pragma unroll
    for (int off = 16; off; off >>= 1) d += __shfl_down(d, off, 32);
    if (threadIdx.x == 0) {
        ws[0] = 0.0f;        // zero accumulator every call (graph replay safe)
        ws[1] = d / (float)B;
    }
}

// err(b,n) = || (M - I) * p_xyz ||, reduced into ws[0].
// B operand (32-bit 4x16, mirrors documented 16x4 A layout):
//   vgpr0: lanes0-15 = K0 (x), lanes16-31 = K2 (z)
//   vgpr1: lanes0-15 = K1 (y), lanes16-31 = K3 (0)
__global__ __launch_bounds__(256) void pc_err_kernel(const float* __restrict__ pc,
                                                     const float* __restrict__ ws_m,
                                                     float* __restrict__ acc,
                                                     int N, int blocksPerBatch) {
    const int b      = blockIdx.x / blocksPerBatch;
    const int lane   = threadIdx.x & 31;
    const int r      = lane & 15;
    const bool hi    = lane >= 16;
    const int wavesPerBlock = blockDim.x >> 5;
    const int wid    = (blockIdx.x % blocksPerBatch) * wavesPerBlock + (threadIdx.x >> 5);
    const int nWaves = blocksPerBatch * wavesPerBlock;

    // A operand: rows 0-2 = (M - I), rows 3-15 = 0.
    // lane L<16 holds A[L][0],A[L][1]; lane L+16 holds A[L][2],A[L][3].
    const float* mb = ws_m + b * 16;
    v2f a; a.x = 0.0f; a.y = 0.0f;
    if (r < 3) {
        if (!hi) { a.x = mb[r * 3 + 0]; a.y = mb[r * 3 + 1]; }
        else     { a.x = mb[r * 3 + 2]; }
    }
    // reconverged here: EXEC all-ones for every WMMA below

    const size_t rowOff = (size_t)b * 4u * (size_t)N;
    const float* px  = pc + rowOff;
    const float* py  = px + N;
    const float* pz  = py + N;
    const float* p02 = hi ? pz : px;        // per-lane base for B vgpr0

    v8f czero = {};
    float accl = 0.0f;

    const int tiles     = N >> 4;           // 16 points per WMMA tile
    const int iters     = tiles / nWaves;   // scalar (from kernel args) -> s_branch loops
    const int mainIters = iters & ~3;

    int t = wid;
    // -------- main loop: 4 tiles / iteration, uniform scalar trip count --------
    for (int i = 0; i < mainIters; i += 4) {
        int n0 = ((t + 0 * nWaves) << 4) + r;
        int n1 = ((t + 1 * nWaves) << 4) + r;
        int n2 = ((t + 2 * nWaves) << 4) + r;
        int n3 = ((t + 3 * nWaves) << 4) + r;
        // issue all 8 loads up front for MLP
        float x0 = p02[n0]; float y0v = py[n0];
        float x1 = p02[n1]; float y1v = py[n1];
        float x2 = p02[n2]; float y2v = py[n2];
        float x3 = p02[n3]; float y3v = py[n3];

        v2f b0; b0.x = x0; b0.y = hi ? 0.0f : y0v;
        v2f b1; b1.x = x1; b1.y = hi ? 0.0f : y1v;
        v2f b2; b2.x = x2; b2.y = hi ? 0.0f : y2v;
        v2f b3; b3.x = x3; b3.y = hi ? 0.0f : y3v;

        v8f d0 = __builtin_amdgcn_wmma_f32_16x16x4_f32(false, a, false, b0, (short)0, czero, false, false);
        v8f d1 = __builtin_amdgcn_wmma_f32_16x16x4_f32(false, a, false, b1, (short)0, czero, false, false);
        v8f d2 = __builtin_amdgcn_wmma_f32_16x16x4_f32(false, a, false, b2, (short)0, czero, false, false);
        v8f d3 = __builtin_amdgcn_wmma_f32_16x16x4_f32(false, a, false, b3, (short)0, czero, false, false);

        accl += sqrtf(d0[0] * d0[0] + d0[1] * d0[1] + d0[2] * d0[2]);
        accl += sqrtf(d1[0] * d1[0] + d1[1] * d1[1] + d1[2] * d1[2]);
        accl += sqrtf(d2[0] * d2[0] + d2[1] * d2[1] + d2[2] * d2[2]);
        accl += sqrtf(d3[0] * d3[0] + d3[1] * d3[1] + d3[2] * d3[2]);
        t += 4 * nWaves;
    }
    // -------- leftover full tiles (iters % 4), still uniform --------
    for (int i = mainIters; i < iters; ++i) {
        int n0 = (t << 4) + r;
        v2f b0; b0.x = p02[n0];
        float yv = py[n0];
        b0.y = hi ? 0.0f : yv;
        v8f d0 = __builtin_amdgcn_wmma_f32_16x16x4_f32(false, a, false, b0, (short)0, czero, false, false);
        accl += sqrtf(d0[0] * d0[0] + d0[1] * d0[1] + d0[2] * d0[2]);
        t += nWaves;
    }
    // -------- tail points (N % (16*nWaves)): pure VALU, may diverge --------
    int covered = iters * nWaves * 16;
    if (covered < N) {
        float m00 = mb[0], m01 = mb[1], m02 = mb[2];
        float m10 = mb[3], m11 = mb[4], m12 = mb[5];
        float m20 = mb[6], m21 = mb[7], m22 = mb[8];
        for (int n = covered + wid * 32 + lane; n < N; n += nWaves * 32) {
            float x = px[n], y = py[n], z = pz[n];
            float dx = fmaf(m00, x, fmaf(m01, y, m02 * z));
            float dy = fmaf(m10, x, fmaf(m11, y, m12 * z));
            float dz = fmaf(m20, x, fmaf(m21, y, m22 * z));
            accl += sqrtf(dx * dx + dy * dy + dz * dz);
        }
    }

#pragma unroll
    for (int off = 16; off; off >>= 1) accl += __shfl_down(accl, off, 32);
    if (lane == 0) atomicAdd(acc, accl);
}

__global__ void finalize_kernel(const float* __restrict__ ws, float* __restrict__ out,
                                int N, int B) {
    float sum = ws[0];
    float lr  = ws[1];
    float pcB = sum / ((float)N * (float)B);   // (sum_b mean_n err) / B
    out[0] = (1.0f - WEIGHT_PC) * (RESCALE_ROT * lr) + WEIGHT_PC * pcB;
    out[1] = lr;
    out[2] = pcB;
}

extern "C" void kernel_launch(void* const* d_in, const int* in_sizes, int n_in,
                              void* d_out, int out_size, void* d_ws, size_t ws_size,
                              hipStream_t stream) {
    const float* pc         = (const float*)d_in[0];
    // d_in[1] = target_transl: unused — translation cancels in M = Rp^T * Rt
    const float* target_rot = (const float*)d_in[2];
    const float* rot_err    = (const float*)d_in[3];
    float* out = (float*)d_out;
    float* ws  = (float*)d_ws;

    const int B = in_sizes[2] / 4;
    const int N = (int)((long long)in_sizes[0] / (4LL * (long long)B));

    prep_kernel<<<1, 32, 0, stream>>>(target_rot, rot_err, ws, B);

    const int blocksPerBatch = 32;             // 256 waves/batch -> 32 tiles/wave
    pc_err_kernel<<<dim3(B * blocksPerBatch), dim3(256), 0, stream>>>(
        pc, ws + 16, ws, N, blocksPerBatch);

    finalize_kernel<<<1, 1, 0, stream>>>(ws, out, N, B);
}